// MyConv2D_Cleaner_7292854468612
// MI455X (gfx1250) — compile-verified
//
#include <hip/hip_runtime.h>

typedef __attribute__((ext_vector_type(2))) float v2f;
typedef __attribute__((ext_vector_type(4))) float v4f;
typedef __attribute__((ext_vector_type(8))) float v8f;

#define P_      23250      // 62*375 patch positions
#define N_      16
#define K_      125        // flattened kernel size
#define KP      128        // K padded to multiple of 4 (WMMA k-step)
#define F_      64
#define PITCH   132        // LDS row pitch in dwords: 128 + 4 -> 16B-aligned rows, 4-bank stride
#define PTILES  ((P_ + 15) / 16)   // 1454

// gfx1250 async copy: global -> LDS, bypasses VGPRs, tracked by ASYNCcnt.
// GVS addressing: SGPR64 base + per-lane 32-bit byte offset; vdst = per-lane LDS byte address.
__device__ __forceinline__ void async_ld_b32_nt(unsigned lds_byte, unsigned glb_byte,
                                                const float* base) {
    asm volatile("global_load_async_to_lds_b32 %0, %1, %2 th:TH_LOAD_NT"
                 :: "v"(lds_byte), "v"(glb_byte), "s"(base));
}

__global__ __launch_bounds__(64)
void conv_cleaner_wmma(const float* __restrict__ x,
                       const float* __restrict__ enc,
                       const float* __restrict__ w,
                       float* __restrict__ out)
{
    __shared__ float klds[F_ * PITCH];       // kernel[f][k], k padded with 0
    __shared__ float elds[16 * PITCH];       // enc tile [c][k]
    __shared__ float xlds[2][16 * PITCH];    // per-wave x tile [c][k]

    const int t    = threadIdx.x;            // 0..63
    const int lane = t & 31;
    const int wave = t >> 5;                 // 0..1
    const int p0   = blockIdx.x * 16;        // p-tile base
    const int ng   = blockIdx.y;             // n-group 0..3

    // ---- zero the K-pad region of the x tiles once (async loads only touch k < 125)
    for (int idx = t; idx < 32 * (KP - K_); idx += 64) {
        int r = idx / (KP - K_);             // 32 rows across both wave buffers
        int k = K_ + idx % (KP - K_);
        (&xlds[0][0])[r * PITCH + k] = 0.0f;
    }

    // ---- stage kernel into LDS: klds[f][k] = w_flat[k*F + f]  (w is [1,5,25,F])
    for (int k = 0; k < KP; ++k) {
        float v = (k < K_) ? w[k * F_ + t] : 0.0f;
        klds[t * PITCH + k] = v;
    }

    // ---- stage enc tile (shared by both waves / all n): elds[c][k] = enc[(p0+c)*K + k]
    for (int it = 0; it < 32; ++it) {
        int idx = it * 64 + t;
        int c   = idx >> 7;                  // 16 rows x 128 k
        int k   = idx & 127;
        int p   = p0 + c; if (p >= P_) p = P_ - 1;   // clamp boundary tile
        float v = (k < K_) ? enc[p * K_ + k] : 0.0f;
        elds[c * PITCH + k] = v;
    }
    __syncthreads();

    const int c = lane & 15;                 // column within tile (== M row for A-frag)
    const int h = lane >> 4;                 // K-half selector (lanes 16-31 carry K=2,3)
    float* xl = &xlds[wave][0];
    const unsigned xl_base = (unsigned)(uintptr_t)xl;   // low 32 bits of flat LDS ptr == LDS offset

    // each wave handles 2 batch indices sequentially (amortizes kernel/enc staging)
    for (int i = 0; i < 2; ++i) {
        const int n = ng * 4 + wave * 2 + i;
        const float* xsrc = x + (size_t)n * P_ * K_;   // wave-uniform -> SGPR base

        // async LDS writes are unordered vs in-flight DS reads of the previous tile
        asm volatile("s_wait_dscnt 0" ::: "memory");

        // ---- async-stage this wave's x tile: xl[c][k] = x[n, p0+c, k] (streamed -> NT)
        for (int it = 0; it < 64; ++it) {
            int idx = it * 32 + lane;
            int cc  = idx >> 7;
            int k   = idx & 127;
            int p   = p0 + cc; if (p >= P_) p = P_ - 1;
            if (k < K_) {
                unsigned goff = (unsigned)((p * K_ + k) * 4);
                unsigned loff = xl_base + (unsigned)((cc * PITCH + k) * 4);
                async_ld_b32_nt(loff, goff, xsrc);
            }
        }
        asm volatile("s_wait_asynccnt 0" ::: "memory");

        // ---- e[col] = <enc[p], x[n,p]> ; lane L does k-half h, b128 LDS reads
        v4f eacc = {0.f, 0.f, 0.f, 0.f};
        #pragma unroll
        for (int kk = 0; kk < 16; ++kk) {
            int off = c * PITCH + h * 64 + kk * 4;      // 16B aligned
            v4f xv = *(const v4f*)(xl + off);
            v4f ev = *(const v4f*)(elds + off);
            eacc += xv * ev;
        }
        float e = eacc.x + eacc.y + eacc.z + eacc.w;
        e += __shfl_xor(e, 16, 32);          // combine the two k-halves; lane L now has e[c]

        // C/D layout: lane = column for every accumulator VGPR -> broadcast init adds e per column
        v8f acc0 = {e, e, e, e, e, e, e, e};
        v8f acc1 = acc0, acc2 = acc0, acc3 = acc0;

        // ---- main GEMM: D[64f x 16p] += kernel[64x128] * xT[128x16], K-steps of 4
        #pragma unroll 4
        for (int kk = 0; kk < 32; ++kk) {
            int ko = kk * 4 + 2 * h;         // this lane's two K elements (ISA 16x16x4 layout)
            v2f b  = *(const v2f*)(xl   + c * PITCH + ko);            // B frag: (K, col c)
            v2f a0 = *(const v2f*)(klds + ( 0 + c) * PITCH + ko);     // A frags: (row f, K)
            v2f a1 = *(const v2f*)(klds + (16 + c) * PITCH + ko);
            v2f a2 = *(const v2f*)(klds + (32 + c) * PITCH + ko);
            v2f a3 = *(const v2f*)(klds + (48 + c) * PITCH + ko);
            acc0 = __builtin_amdgcn_wmma_f32_16x16x4_f32(false, a0, false, b, (short)0, acc0, false, false);
            acc1 = __builtin_amdgcn_wmma_f32_16x16x4_f32(false, a1, false, b, (short)0, acc1, false, false);
            acc2 = __builtin_amdgcn_wmma_f32_16x16x4_f32(false, a2, false, b, (short)0, acc2, false, false);
            acc3 = __builtin_amdgcn_wmma_f32_16x16x4_f32(false, a3, false, b, (short)0, acc3, false, false);
        }

        // ---- store: VGPR v holds rows M=v (lanes 0-15) and M=v+8 (lanes 16-31)
        // out row r = f*16 + n ; streamed once -> non-temporal stores keep enc/kernel in L2
        const int p = p0 + c;
        if (p < P_) {
            #pragma unroll
            for (int v = 0; v < 8; ++v) {
                int fr = v + 8 * h;          // row within the 16-row f-tile
                __builtin_nontemporal_store(acc0[v], &out[((size_t)( 0 + fr) * N_ + n) * P_ + p]);
                __builtin_nontemporal_store(acc1[v], &out[((size_t)(16 + fr) * N_ + n) * P_ + p]);
                __builtin_nontemporal_store(acc2[v], &out[((size_t)(32 + fr) * N_ + n) * P_ + p]);
                __builtin_nontemporal_store(acc3[v], &out[((size_t)(48 + fr) * N_ + n) * P_ + p]);
            }
        }
    }
}

extern "C" void kernel_launch(void* const* d_in, const int* in_sizes, int n_in,
                              void* d_out, int out_size, void* d_ws, size_t ws_size,
                              hipStream_t stream) {
    const float* x   = (const float*)d_in[0];   // [16, 23250, 125]
    const float* enc = (const float*)d_in[1];   // [23250, 125]
    const float* w   = (const float*)d_in[2];   // [1, 5, 25, 64]
    float* out       = (float*)d_out;           // [16, 64, 62, 375] == flat [1024, 23250]

    (void)in_sizes; (void)n_in; (void)out_size; (void)d_ws; (void)ws_size;

    dim3 grid(PTILES, 4);                       // 1454 p-tiles x 4 n-groups
    conv_cleaner_wmma<<<grid, 64, 0, stream>>>(x, enc, w, out);
}